// DotPredictor_12704513261757
// MI455X (gfx1250) — compile-verified
//
#include <hip/hip_runtime.h>

// Edge-wise dot product: score[e] = dot(h[src[e]], h[dst[e]]), D=16.
//
// Roofline: ~0.2 FLOP/byte -> pure memory problem. h (6.4 MB) is L2-resident
// (192 MB L2), so per-edge row gathers are L2 hits; HBM floor is ~70 MB
// (index streams + one pass of h + output) ~= 3 us at 23.3 TB/s.
// Strategy:
//   * one edge per lane, rows fetched as 4x global_load_b128 (64B aligned)
//   * src/dst index streams staged into LDS with global_load_async_to_lds_b32
//     (ASYNCcnt), double-buffered: next tile's indices are in flight while
//     the current tile gathers/computes. Only the low 32 bits of each int64
//     index are fetched (ids < 2^31, little-endian).
//   * all address math in 32 bits; nontemporal output store
//   * uniform per-block fast path: only the final partial block pays for
//     clamps and per-edge store guards (scalar branch, no exec churn)
//   * WMMA deliberately NOT used: diagonal-of-16x16 via WMMA moves the same
//     bytes but needs more cross-lane shuffle instructions than the 16
//     lane-parallel v_fmac_f32 it replaces.

#define BLOCK 256
#define EPT   8   // edges per thread

typedef float f4 __attribute__((ext_vector_type(4)));

// Issue async (ASYNCcnt-tracked) 4-byte index loads into LDS.
// GVS addressing: saddr = uniform int64-array base (SGPR pair),
// vaddr = per-lane signed 32-bit byte offset (e*8 -> low dword of the int64),
// vdst = per-lane LDS byte address (generic-pointer low 32 bits).
__device__ __forceinline__ void issue_idx_pair(
    const long long* src, const long long* dst,
    int* lds_src_slot, int* lds_dst_slot, int e)
{
    unsigned goff  = (unsigned)e << 3;
    unsigned lds_s = (unsigned)(size_t)lds_src_slot;
    unsigned lds_d = (unsigned)(size_t)lds_dst_slot;
    asm volatile("global_load_async_to_lds_b32 %0, %1, %2"
                 :: "v"(lds_s), "v"(goff), "s"(src) : "memory");
    asm volatile("global_load_async_to_lds_b32 %0, %1, %2"
                 :: "v"(lds_d), "v"(goff), "s"(dst) : "memory");
}

template <bool GUARD>
__device__ __forceinline__ void run_tiles(
    const float* __restrict__ h,
    const long long* __restrict__ src,
    const long long* __restrict__ dst,
    float* __restrict__ out,
    int n_edges, int tile_base, int tid,
    int (*s_src)[BLOCK], int (*s_dst)[BLOCK])
{
    auto clamp_e = [&](int e) -> int {
        return GUARD ? (e < n_edges ? e : (n_edges - 1)) : e;
    };

    // Prologue: start tile 0's indices.
    issue_idx_pair(src, dst, &s_src[0][tid], &s_dst[0][tid],
                   clamp_e(tile_base + tid));

    for (int s = 0; s < EPT; ++s) {
        const int buf = s & 1;
        const int e = tile_base + s * BLOCK + tid;

        if (s + 1 < EPT) {
            // Kick off next tile's indices, then wait until only those 2
            // remain outstanding (async loads complete in order) -> current
            // buffer is ready while the next buffer streams in.
            issue_idx_pair(src, dst, &s_src[buf ^ 1][tid], &s_dst[buf ^ 1][tid],
                           clamp_e(tile_base + (s + 1) * BLOCK + tid));
            asm volatile("s_wait_asynccnt 2" ::: "memory");
        } else {
            asm volatile("s_wait_asynccnt 0" ::: "memory");
        }

        const unsigned si = (unsigned)s_src[buf][tid];
        const unsigned di = (unsigned)s_dst[buf][tid];

        // 64B rows, 64B-aligned -> 4x global_load_b128 per row; default
        // temporal hint keeps h hot in L2 (it fits entirely).
        const f4* __restrict__ ps = (const f4*)((const char*)h + (size_t)(si << 6));
        const f4* __restrict__ pd = (const f4*)((const char*)h + (size_t)(di << 6));
        f4 a0 = ps[0], a1 = ps[1], a2 = ps[2], a3 = ps[3];
        f4 b0 = pd[0], b1 = pd[1], b2 = pd[2], b3 = pd[3];

        f4 acc = a0 * b0;
        acc += a1 * b1;
        acc += a2 * b2;
        acc += a3 * b3;
        float score = (acc.x + acc.y) + (acc.z + acc.w);

        if (!GUARD || e < n_edges) {
            __builtin_nontemporal_store(score, &out[e]);  // streamed output
        }
    }
}

__global__ __launch_bounds__(BLOCK) void edge_dot_kernel(
    const float* __restrict__ h,
    const long long* __restrict__ src,
    const long long* __restrict__ dst,
    float* __restrict__ out,
    int n_edges)
{
    // Low 32 bits of each int64 index, double buffered (4 KB total).
    __shared__ int s_src[2][BLOCK];
    __shared__ int s_dst[2][BLOCK];

    const int tid = threadIdx.x;
    const int tile_base = blockIdx.x * (BLOCK * EPT);   // < 2^31

    // Uniform (scalar) branch: all blocks except the last are full tiles.
    if (tile_base + BLOCK * EPT <= n_edges) {
        run_tiles<false>(h, src, dst, out, n_edges, tile_base, tid, s_src, s_dst);
    } else {
        run_tiles<true >(h, src, dst, out, n_edges, tile_base, tid, s_src, s_dst);
    }
}

extern "C" void kernel_launch(void* const* d_in, const int* in_sizes, int n_in,
                              void* d_out, int out_size, void* d_ws, size_t ws_size,
                              hipStream_t stream) {
    const float*     h   = (const float*)d_in[0];
    const long long* src = (const long long*)d_in[1];   // int64 per reference
    const long long* dst = (const long long*)d_in[2];   // int64 per reference
    float*           out = (float*)d_out;

    const int n_edges = in_sizes[1];
    const int tile    = BLOCK * EPT;
    const int blocks  = (n_edges + tile - 1) / tile;

    edge_dot_kernel<<<blocks, BLOCK, 0, stream>>>(h, src, dst, out, n_edges);
}